// HungarianLoss_25202868093039
// MI455X (gfx1250) — compile-verified
//
#include <hip/hip_runtime.h>

// ---------------------------------------------------------------------------
// Hungarian (4x4 assignment) wing-loss mean over N=2M samples.
// Memory-bound streaming reduction: 128 MB in, 4 B out -> ~5.5 us floor @23.3TB/s.
// Strategy: NT b128 loads, unconditional speculative L2 prefetch, register-
// resident 4x4 wing cost + subset-DP assignment, WMMA wave reduction +
// ds_swizzle half-wave combine, deterministic 2-pass block reduce.
// ---------------------------------------------------------------------------

typedef float v2f __attribute__((ext_vector_type(2)));
typedef float v4f __attribute__((ext_vector_type(4)));
typedef float v8f __attribute__((ext_vector_type(8)));

#define OMEGA_F       10.0f
#define NEG_WING_C    7.91759469228055f   // -(OMEGA*(1-ln(1+OMEGA/EPS))) = -(10*(1-ln6))
#define TEN_LN2       6.93147180559945f   // 10*ln(2): 10*log1p(x) == TEN_LN2*log2(1+x)

// wing(d) for d >= 0:  d < 10 ? 10*log1p(d/2) : d - WING_C
__device__ __forceinline__ float wing_loss(float d) {
  // v_log_f32 (log2) + fma; TRANS co-executes with VALU on CDNA5
  float t = TEN_LN2 * __builtin_amdgcn_logf(__builtin_fmaf(0.5f, d, 1.0f));
  return (d < OMEGA_F) ? t : (d + NEG_WING_C);
}

// ---------------------------------------------------------------------------
// 256-thread block sum. Wave32 stage done with V_WMMA_F32_16X16X4_F32:
//   A(16x4): VGPR0 = lane value, VGPR1 = 0  ->  A[m][0]=acc_m, A[m][2]=acc_{m+16}
//   B(4x16): all ones
//   D[m][n] = acc_m + acc_{m+16}  (replicated across n)
// Lane n<16 holds rows 0..7 in D[0..7]; lane n>=16 holds rows 8..15.
// sum(D[0..7]) + ds_swizzle SWAPX16 partner == full 32-lane sum.
// Result valid on thread 0.
// ---------------------------------------------------------------------------
__device__ __forceinline__ float block_reduce_sum_256(float v) {
  v2f A; A.x = v;    A.y = 0.0f;
  v2f B; B.x = 1.0f; B.y = 1.0f;
  v8f C = {0.0f, 0.0f, 0.0f, 0.0f, 0.0f, 0.0f, 0.0f, 0.0f};
  v8f D = __builtin_amdgcn_wmma_f32_16x16x4_f32(
      /*neg_a=*/false, A, /*neg_b=*/false, B,
      /*c_mod=*/(short)0, C, /*reuse_a=*/false, /*reuse_b=*/false);
  float s = ((D[0] + D[1]) + (D[2] + D[3])) + ((D[4] + D[5]) + (D[6] + D[7]));

  // SWAPX16 via ds_swizzle_b32: group-of-32, xor_mask=0x10, and_mask=0x1f
  int sw = __builtin_amdgcn_ds_swizzle(__builtin_bit_cast(int, s), 0x401F);
  s += __builtin_bit_cast(float, sw);

  __shared__ float lds[8];           // 8 waves per 256-thread block
  const int lane = threadIdx.x & 31;
  const int wave = threadIdx.x >> 5;
  if (lane == 0) lds[wave] = s;
  __syncthreads();
  float tot = 0.0f;
  if (threadIdx.x == 0) {
#pragma unroll
    for (int w = 0; w < 8; ++w) tot += lds[w];   // fixed order -> deterministic
  }
  return tot;
}

// ---------------------------------------------------------------------------
// Kernel 1: per-block partial sums of the per-sample matched loss.
// ---------------------------------------------------------------------------
__global__ __launch_bounds__(256) void wing_hungarian_partial(
    const float* __restrict__ pred, const float* __restrict__ targ,
    float* __restrict__ partial, int n) {
  const v4f* __restrict__ pv = (const v4f*)pred;
  const v4f* __restrict__ tv = (const v4f*)targ;
  const int stride = (int)(gridDim.x * blockDim.x);
  float acc = 0.0f;

  for (int i = (int)(blockIdx.x * blockDim.x + threadIdx.x); i < n; i += stride) {
    // 4 x global_load_b128, non-temporal (single-use 128 MB stream)
    v4f pa = __builtin_nontemporal_load(pv + 2 * i);
    v4f pb = __builtin_nontemporal_load(pv + 2 * i + 1);
    v4f ta = __builtin_nontemporal_load(tv + 2 * i);
    v4f tb = __builtin_nontemporal_load(tv + 2 * i + 1);

    // Unconditional speculative prefetch of the next grid-stride tile toward
    // L2 (global_prefetch_b8). Speculative prefetch translation failures are
    // silently dropped (ISA 10.5), so no bounds check / exec-mask churn needed.
    __builtin_prefetch((const void*)(pv + 2 * (i + stride)), 0, 0);
    __builtin_prefetch((const void*)(tv + 2 * (i + stride)), 0, 0);

    const float px[4] = {pa.x, pa.z, pb.x, pb.z};
    const float py[4] = {pa.y, pa.w, pb.y, pb.w};
    const float tx[4] = {ta.x, ta.z, tb.x, tb.z};
    const float ty[4] = {ta.y, ta.w, tb.y, tb.w};

    // cost[r][j] = wing(|px_j - tx_r|) + wing(|py_j - ty_r|)   (32 v_log_f32)
    float c[4][4];
#pragma unroll
    for (int r = 0; r < 4; ++r) {
#pragma unroll
      for (int j = 0; j < 4; ++j) {
        c[r][j] = wing_loss(__builtin_fabsf(px[j] - tx[r])) +
                  wing_loss(__builtin_fabsf(py[j] - ty[r]));
      }
    }

    // 4x4 assignment problem via subset DP (== min over all 24 permutations):
    // dp[mask] = min cost assigning rows 0..popcnt(mask)-1 to columns in mask.
    // 32 fused add+min ops, fully unrolled, all register-indexed.
    float dp[16];
    dp[0] = 0.0f;
#pragma unroll
    for (int mask = 1; mask < 16; ++mask) {
      const int row = __popc(mask) - 1;      // constant-folds after unroll
      float best = 3.402823466e38f;
#pragma unroll
      for (int j = 0; j < 4; ++j) {
        if (mask & (1 << j)) {
          best = fminf(best, dp[mask & ~(1 << j)] + c[row][j]);
        }
      }
      dp[mask] = best;
    }
    acc += dp[15];
  }

  const float tot = block_reduce_sum_256(acc);   // EXEC all-ones here
  if (threadIdx.x == 0) partial[blockIdx.x] = tot;
}

// ---------------------------------------------------------------------------
// Kernel 2: deterministic finish — fixed-order sum of block partials, * 1/N.
// ---------------------------------------------------------------------------
__global__ __launch_bounds__(256) void wing_hungarian_finish(
    const float* __restrict__ partial, float* __restrict__ out,
    int nb, float inv_n) {
  float acc = 0.0f;
  for (int i = (int)threadIdx.x; i < nb; i += 256) acc += partial[i];
  const float tot = block_reduce_sum_256(acc);
  if (threadIdx.x == 0) out[0] = tot * inv_n;
}

// ---------------------------------------------------------------------------
extern "C" void kernel_launch(void* const* d_in, const int* in_sizes, int n_in,
                              void* d_out, int out_size, void* d_ws, size_t ws_size,
                              hipStream_t stream) {
  const float* pred = (const float*)d_in[0];
  const float* targ = (const float*)d_in[1];
  float* out        = (float*)d_out;
  float* partial    = (float*)d_ws;

  const int n = in_sizes[0] / 8;               // samples (quads)

  int blocks = (n + 255) / 256;
  if (blocks > 2048) blocks = 2048;            // ~524K threads, ~4 samples/thread
  const size_t max_partials = ws_size / sizeof(float);
  if ((size_t)blocks > max_partials) blocks = (int)max_partials;
  if (blocks < 1) blocks = 1;

  wing_hungarian_partial<<<blocks, 256, 0, stream>>>(pred, targ, partial, n);
  wing_hungarian_finish<<<1, 256, 0, stream>>>(partial, out, blocks,
                                               1.0f / (float)n);
}